// GraphSAGE_13993003450942
// MI455X (gfx1250) — compile-verified
//
#include <hip/hip_runtime.h>
#include <hip/hip_bf16.h>

typedef __attribute__((ext_vector_type(2))) float v2f;
typedef __attribute__((ext_vector_type(8))) float v8f;

#define NNODES 50000
#define NEDGES 800000
#define NPAIRS 500000
#define F_IN   128
#define HID    256
#define EIN    16
#define ZK     (2 * HID + EIN)   // 528
#define GN     4                 // n-tiles per wave in SAGE GEMM (16/4 groups)
#define ET     5                 // edge tiles per wave in edge MLP (31250/5 exact)

// D = A(16x4) * B(4x16) + C, exact fp32 accumulate.
// A fragment: lane m = L%16; VGPR0 holds K=koff, VGPR1 K=koff+1
// (koff = 0 for lanes 0-15, 2 for lanes 16-31). B fragment uses the same
// (koff, koff+1) addressing with n = L%16.
__device__ __forceinline__ v8f wmma_ab(v2f a, v2f b, v8f c) {
  return __builtin_amdgcn_wmma_f32_16x16x4_f32(false, a, false, b,
                                               (short)0, c, false, false);
}

__device__ __forceinline__ v2f ld2(const float* __restrict__ p, int k) {
  v2f r;
  r.x = p[k];
  r.y = p[k + 1];
  return r;   // merges into global_load_b64
}

__global__ void k_zero(float* __restrict__ p, int n) {
  int i = blockIdx.x * blockDim.x + threadIdx.x;
  if (i < n) p[i] = 0.f;
}

__global__ void k_count(const int* __restrict__ dst, float* __restrict__ cnt) {
  int e = blockIdx.x * blockDim.x + threadIdx.x;
  if (e < NEDGES) atomicAdd(&cnt[dst[e]], 1.f);
}

// edge-parallel scatter-add: agg[dst[e]][f] += x[src[e]][f], F = 1<<lgF
__global__ void k_scatter(const int* __restrict__ src, const int* __restrict__ dst,
                          const float* __restrict__ x, float* __restrict__ agg, int lgF) {
  int i = blockIdx.x * blockDim.x + threadIdx.x;
  int tot = NEDGES << lgF;
  if (i >= tot) return;
  int e = i >> lgF;
  int f = i & ((1 << lgF) - 1);
  atomicAdd(&agg[(dst[e] << lgF) + f], x[(src[e] << lgF) + f]);
}

__global__ void k_norm(float* __restrict__ agg, const float* __restrict__ cnt, int lgF) {
  int i = blockIdx.x * blockDim.x + threadIdx.x;
  int tot = NNODES << lgF;
  if (i >= tot) return;
  int n = i >> lgF;
  agg[i] *= 1.f / fmaxf(cnt[n], 1.f);
}

// out = relu(A @ Wl^T + bias + X @ Wr^T), A,X: [NNODES,K], Wl,Wr: [HID,K]
// One wave per 16-row x 64-col block: 4 independent 16x16 WMMA accumulator
// chains sharing the A/X fragments (4x less A traffic, hides XDL latency).
__global__ void k_sage_gemm(const float* __restrict__ A, const float* __restrict__ X,
                            const float* __restrict__ Wl, const float* __restrict__ Wr,
                            const float* __restrict__ bias, float* __restrict__ out,
                            int K) {
  int gtid = blockIdx.x * blockDim.x + threadIdx.x;
  int wave = gtid >> 5;
  int lane = threadIdx.x & 31;
  const int ngrp = (HID / 16) / GN;        // 4 n-groups
  const int tiles_m = NNODES / 16;         // 3125
  if (wave >= tiles_m * ngrp) return;      // wave-uniform guard
  int tm = wave / ngrp;
  int tg = wave - tm * ngrp;
  int koff = (lane >> 4) << 1;             // 0 or 2
  int row = tm * 16 + (lane & 15);
  int col0 = tg * (GN * 16) + (lane & 15);
  const float* arow = A + row * K;
  const float* xrow = X + row * K;
  const float* wl[GN];
  const float* wr[GN];
#pragma unroll
  for (int t = 0; t < GN; ++t) {
    int col = col0 + t * 16;
    wl[t] = Wl + col * K;
    wr[t] = Wr + col * K;
  }
  v8f c[GN] = {};
  for (int kb = 0; kb < K; kb += 4) {
    int k = kb + koff;
    v2f a = ld2(arow, k);
    v2f x2 = ld2(xrow, k);
#pragma unroll
    for (int t = 0; t < GN; ++t) {
      c[t] = wmma_ab(a, ld2(wl[t], k), c[t]);
      c[t] = wmma_ab(x2, ld2(wr[t], k), c[t]);
    }
  }
  int mbase = tm * 16 + ((lane >> 4) << 3);   // C/D: VGPR r -> M=r / r+8
#pragma unroll
  for (int t = 0; t < GN; ++t) {
    int col = col0 + t * 16;
    float bv = bias[col];
#pragma unroll
    for (int r = 0; r < 8; ++r)
      out[(mbase + r) * HID + col] = fmaxf(c[t][r] + bv, 0.f);
  }
}

// Fused edge MLP: z = [H[u], H[v], eattr] (528), s = relu(z@W1^T+b1)@W2^T+b2
// out[p] = log_exposure[p] + s. One wave per 5 edge tiles (80 edges): each
// W1 B-fragment is loaded once and fed to 5 independent accumulator chains
// (5x less W1/L2 traffic, back-to-back WMMA issue).
__global__ void k_edge_mlp(const float* __restrict__ H, const int* __restrict__ eu,
                           const int* __restrict__ ev, const float* __restrict__ eattr,
                           const float* __restrict__ logexp,
                           const float* __restrict__ W1, const float* __restrict__ b1,
                           const float* __restrict__ W2, const float* __restrict__ b2,
                           float* __restrict__ out) {
  int gtid = blockIdx.x * blockDim.x + threadIdx.x;
  int wave = gtid >> 5;
  int lane = threadIdx.x & 31;
  const int groups = NPAIRS / 16 / ET;     // 6250
  if (wave >= groups) return;              // wave-uniform guard
  int koff = (lane >> 4) << 1;
  int p0 = wave * (ET * 16) + (lane & 15);
  const float* urow[ET];
  const float* vrow[ET];
  const float* arow[ET];
#pragma unroll
  for (int t = 0; t < ET; ++t) {
    int p = p0 + t * 16;
    urow[t] = H + eu[p] * HID;
    vrow[t] = H + ev[p] * HID;
    arow[t] = eattr + p * EIN;
  }
  float acc[ET][8] = {};
  float b2v = b2[0];

  for (int nt = 0; nt < HID / 16; ++nt) {
    int col = nt * 16 + (lane & 15);
    const float* w = W1 + col * ZK;
    v8f c[ET] = {};
    // K segment 1: h[u] columns 0..255
    for (int kb = 0; kb < HID; kb += 4) {
      v2f b = ld2(w, kb + koff);
#pragma unroll
      for (int t = 0; t < ET; ++t)
        c[t] = wmma_ab(ld2(urow[t], kb + koff), b, c[t]);
    }
    // K segment 2: h[v] columns 256..511
    for (int kb = 0; kb < HID; kb += 4) {
      v2f b = ld2(w, HID + kb + koff);
#pragma unroll
      for (int t = 0; t < ET; ++t)
        c[t] = wmma_ab(ld2(vrow[t], kb + koff), b, c[t]);
    }
    // K segment 3: edge_attr columns 512..527
#pragma unroll
    for (int kb = 0; kb < EIN; kb += 4) {
      v2f b = ld2(w, 2 * HID + kb + koff);
#pragma unroll
      for (int t = 0; t < ET; ++t)
        c[t] = wmma_ab(ld2(arow[t], kb + koff), b, c[t]);
    }
    float bv = b1[col];
    float wv = W2[col];
#pragma unroll
    for (int t = 0; t < ET; ++t)
#pragma unroll
      for (int r = 0; r < 8; ++r)
        acc[t][r] += fmaxf(c[t][r] + bv, 0.f) * wv;
  }

  // Reduce across the 16-lane half (xor masks 1,2,4,8 stay within a half).
#pragma unroll
  for (int t = 0; t < ET; ++t) {
#pragma unroll
    for (int r = 0; r < 8; ++r) {
      float s = acc[t][r];
      s += __shfl_xor(s, 1);
      s += __shfl_xor(s, 2);
      s += __shfl_xor(s, 4);
      s += __shfl_xor(s, 8);
      acc[t][r] = s;
    }
  }
  if ((lane & 15) == 0) {
    int half = (lane >> 4) << 3;   // lane 0 -> edges 0..7, lane 16 -> 8..15
#pragma unroll
    for (int t = 0; t < ET; ++t) {
      int base = (wave * ET + t) * 16 + half;
#pragma unroll
      for (int r = 0; r < 8; ++r)
        out[base + r] = logexp[base + r] + acc[t][r] + b2v;
    }
  }
}

extern "C" void kernel_launch(void* const* d_in, const int* in_sizes, int n_in,
                              void* d_out, int out_size, void* d_ws, size_t ws_size,
                              hipStream_t stream) {
  (void)in_sizes; (void)n_in; (void)out_size; (void)ws_size;
  const float* x      = (const float*)d_in[0];
  const int*   eidx   = (const int*)d_in[1];
  const int*   eu     = (const int*)d_in[2];
  const int*   ev     = (const int*)d_in[3];
  const float* eattr  = (const float*)d_in[4];
  const float* logexp = (const float*)d_in[5];
  const float* Wl1    = (const float*)d_in[6];
  const float* bl1    = (const float*)d_in[7];
  const float* Wr1    = (const float*)d_in[8];
  const float* Wl2    = (const float*)d_in[9];
  const float* bl2    = (const float*)d_in[10];
  const float* Wr2    = (const float*)d_in[11];
  const float* W1     = (const float*)d_in[12];
  const float* b1     = (const float*)d_in[13];
  const float* W2     = (const float*)d_in[14];
  const float* b2     = (const float*)d_in[15];
  float* out = (float*)d_out;

  const int* src = eidx;            // edge_index[0]
  const int* dst = eidx + NEDGES;   // edge_index[1]

  float* ws  = (float*)d_ws;
  float* cnt = ws;                  // [N]
  float* agg = ws + NNODES;         // [N,256] (conv1 uses [N,128])
  float* h1  = agg + NNODES * HID;  // [N,256]
  float* h2  = h1 + NNODES * HID;   // [N,256]

  const int TB = 256;
  int gemm_waves = (NNODES / 16) * ((HID / 16) / GN);   // 12500 waves
  int gemm_blocks = (gemm_waves * 32 + TB - 1) / TB;    // 1563 (guarded)

  // ---- conv1 ----
  {
    int nz = NNODES * (1 + F_IN);   // cnt + agg(128) contiguous
    k_zero<<<(nz + TB - 1) / TB, TB, 0, stream>>>(cnt, nz);
    k_count<<<(NEDGES + TB - 1) / TB, TB, 0, stream>>>(dst, cnt);
    int nt = NEDGES << 7;
    k_scatter<<<(nt + TB - 1) / TB, TB, 0, stream>>>(src, dst, x, agg, 7);
    int nn = NNODES << 7;
    k_norm<<<(nn + TB - 1) / TB, TB, 0, stream>>>(agg, cnt, 7);
    k_sage_gemm<<<gemm_blocks, TB, 0, stream>>>(agg, x, Wl1, Wr1, bl1, h1, F_IN);
  }
  // ---- conv2 ----
  {
    int nz = NNODES * HID;
    k_zero<<<(nz + TB - 1) / TB, TB, 0, stream>>>(agg, nz);
    int nt = NEDGES << 8;
    k_scatter<<<(nt + TB - 1) / TB, TB, 0, stream>>>(src, dst, h1, agg, 8);
    int nn = NNODES << 8;
    k_norm<<<(nn + TB - 1) / TB, TB, 0, stream>>>(agg, cnt, 8);
    k_sage_gemm<<<gemm_blocks, TB, 0, stream>>>(agg, h1, Wl2, Wr2, bl2, h2, HID);
  }
  // ---- edge MLP ----
  {
    int waves = NPAIRS / 16 / ET;                     // 6250
    int blocks = (waves * 32 + TB - 1) / TB;          // 782 (guarded)
    k_edge_mlp<<<blocks, TB, 0, stream>>>(h2, eu, ev, eattr, logexp,
                                          W1, b1, W2, b2, out);
  }
}